// SVConv_Brute_23424751633061
// MI455X (gfx1250) — compile-verified
//
#include <hip/hip_runtime.h>
#include <hip/hip_bf16.h>
#include <stdint.h>

// Problem constants (match reference)
#define BATCH   2
#define CHN     3
#define HH      512
#define WW      512
#define KH      7
#define KW      7
#define KTAPS   49      // KH*KW
#define PADR    3       // reflect pad
#define TILE    128     // output pixels per block (one row strip)
#define THREADS 256     // 8 wave32 waves; each wave owns 16 pixels (WMMA group)
#define XCOLS   (TILE + KW - 1)   // 134 staged x columns

typedef float v2f __attribute__((ext_vector_type(2)));
typedef float v8f __attribute__((ext_vector_type(8)));

__device__ __forceinline__ int reflect_idx(int v, int n) {
    v = (v < 0) ? -v : v;
    v = (v >= n) ? (2 * n - 2 - v) : v;
    return v;
}

// ---- CDNA5 async global->LDS copies (ASYNCcnt). GVS mode: saddr + vaddr ----
__device__ __forceinline__ void async_ld_b128(uint32_t lds_off, uint64_t sbase, uint32_t goff) {
    asm volatile("global_load_async_to_lds_b128 %0, %1, %2"
                 :: "v"(lds_off), "v"(goff), "s"(sbase) : "memory");
}
__device__ __forceinline__ void async_ld_b32(uint32_t lds_off, uint64_t sbase, uint32_t goff) {
    asm volatile("global_load_async_to_lds_b32 %0, %1, %2"
                 :: "v"(lds_off), "v"(goff), "s"(sbase) : "memory");
}
__device__ __forceinline__ void wait_async0() {
#if __has_builtin(__builtin_amdgcn_s_wait_asynccnt)
    __builtin_amdgcn_s_wait_asynccnt(0);
#else
    asm volatile("s_wait_asynccnt 0" ::: "memory");
#endif
}

__global__ __launch_bounds__(THREADS) void svconv_psf_kernel(
    const float* __restrict__ x, const float* __restrict__ kern, float* __restrict__ out)
{
    __shared__ __align__(16) float lds_k[TILE * KTAPS];          // 25088 B
    __shared__ __align__(16) float lds_x[BATCH][KH][XCOLS];      //  7504 B

    const int t  = threadIdx.x;
    const int j0 = blockIdx.x * TILE;
    const int i  = blockIdx.y;
    const int c  = blockIdx.z;

    // ---- stage weight tile: contiguous TILE*49 floats, coalesced 16B/lane ----
    const uint64_t kbase = (uint64_t)(kern + ((size_t)(c * HH + i) * WW + j0) * KTAPS);
    const uint32_t lds_k_base = (uint32_t)(uintptr_t)&lds_k[0];
    const int nchunks = (TILE * KTAPS * 4) / 16;   // 1568 x 16B
    for (int ch = t; ch < nchunks; ch += THREADS)
        async_ld_b128(lds_k_base + (uint32_t)ch * 16u, kbase, (uint32_t)ch * 16u);

    // ---- stage x window (reflect applied at staging time) ----
    const uint64_t xbase = (uint64_t)x;
    const uint32_t lds_x_base = (uint32_t)(uintptr_t)&lds_x[0][0][0];
    const int xn = BATCH * KH * XCOLS;             // 1876 elements
    for (int idx = t; idx < xn; idx += THREADS) {
        int col = idx % XCOLS;
        int rem = idx / XCOLS;
        int p   = rem % KH;
        int b   = rem / KH;
        int gr  = reflect_idx(i - PADR + p, HH);
        int gc  = reflect_idx(j0 - PADR + col, WW);
        uint32_t goff = (uint32_t)((((b * CHN + c) * HH + gr) * WW + gc) * 4);
        async_ld_b32(lds_x_base + (uint32_t)idx * 4u, xbase, goff);
    }

    wait_async0();
    __syncthreads();

    const int lane = t & 31;
    const int wave = t >> 5;
    const int jloc = wave * 16;        // this wave's 16-pixel group within tile

#if __has_builtin(__builtin_amdgcn_wmma_f32_16x16x4_f32)
    // A[16x4]: lane m=lane&15 holds pixel row; lane halves hold K pairs {0,1}/{2,3}.
    // B[4x16]: mirrored layout, column n = lane&15. Accumulate 13 K-chunks (49 taps, zero-padded).
    const int m  = lane & 15;
    const int ks = (lane >> 4) * 2;
    v8f acc0 = {};
    v8f acc1 = {};
    #pragma unroll
    for (int k0 = 0; k0 < 52; k0 += 4) {
        const int ka0 = k0 + ks;
        const int ka1 = ka0 + 1;
        const int s0  = (ka0 < KTAPS) ? ka0 : 0;
        const int s1  = (ka1 < KTAPS) ? ka1 : 0;
        const int p0 = s0 / KW, q0 = s0 % KW;
        const int p1 = s1 / KW, q1 = s1 % KW;
        v2f a, b0, b1;
        a.x  = (ka0 < KTAPS) ? lds_k[(jloc + m) * KTAPS + s0] : 0.0f;
        a.y  = (ka1 < KTAPS) ? lds_k[(jloc + m) * KTAPS + s1] : 0.0f;
        b0.x = lds_x[0][p0][jloc + m + q0];
        b0.y = lds_x[0][p1][jloc + m + q1];
        b1.x = lds_x[1][p0][jloc + m + q0];
        b1.y = lds_x[1][p1][jloc + m + q1];
        acc0 = __builtin_amdgcn_wmma_f32_16x16x4_f32(false, a, false, b0,
                                                     (short)0, acc0, false, false);
        acc1 = __builtin_amdgcn_wmma_f32_16x16x4_f32(false, a, false, b1,
                                                     (short)0, acc1, false, false);
    }
    // Diagonal extraction: D[j,j] lives at VGPR (j&7), lane (j<8 ? j : j+16).
    // Lanes 0-15 emit batch 0, lanes 16-31 emit batch 1.
    const int half = lane >> 4;
    const int src  = (m < 8) ? m : (m + 16);
    float ov = 0.0f;
    #pragma unroll
    for (int r = 0; r < 8; ++r) {
        float t0 = __shfl(acc0[r], src, 32);
        float t1 = __shfl(acc1[r], src, 32);
        if ((m & 7) == r) ov = half ? t1 : t0;
    }
    out[((size_t)(half * CHN + c) * HH + i) * WW + j0 + jloc + m] = ov;
#else
    // Scalar FMA fallback: thread (b = t>>7, j = t&127) does one pixel for one batch.
    const int jf = t & (TILE - 1);
    const int bf = t >> 7;
    float acc = 0.0f;
    #pragma unroll
    for (int p = 0; p < KH; ++p) {
        #pragma unroll
        for (int q = 0; q < KW; ++q)
            acc = fmaf(lds_k[jf * KTAPS + p * KW + q], lds_x[bf][p][jf + q], acc);
    }
    out[((size_t)(bf * CHN + c) * HH + i) * WW + j0 + jf] = acc;
    (void)lane; (void)wave; (void)jloc;
#endif
}

extern "C" void kernel_launch(void* const* d_in, const int* in_sizes, int n_in,
                              void* d_out, int out_size, void* d_ws, size_t ws_size,
                              hipStream_t stream) {
    (void)in_sizes; (void)n_in; (void)d_ws; (void)ws_size; (void)out_size;
    const float* x    = (const float*)d_in[0];
    const float* kern = (const float*)d_in[1];
    float* out        = (float*)d_out;
    dim3 grid(WW / TILE, HH, CHN);   // 4 x 512 x 3 = 6144 blocks
    svconv_psf_kernel<<<grid, dim3(THREADS), 0, stream>>>(x, kern, out);
}